// diff_pool_net_61229053771786
// MI455X (gfx1250) — compile-verified
//
#include <hip/hip_runtime.h>
#include <hip/hip_bf16.h>
#include <math.h>

#define NN 6000
#define EE 192000

typedef float v8f __attribute__((ext_vector_type(8)));
typedef float v2f __attribute__((ext_vector_type(2)));

// ---------------- CSR build ----------------
__global__ void zero_i32_kernel(int* p, int n) {
  int i = blockIdx.x * blockDim.x + threadIdx.x;
  if (i < n) p[i] = 0;
}

__global__ void edge_count_kernel(const int* __restrict__ e0, int* __restrict__ cnt) {
  int i = blockIdx.x * blockDim.x + threadIdx.x;
  if (i < EE) atomicAdd(&cnt[e0[i]], 1);
}

// single-workgroup exclusive scan over n counts -> row_ptr[0..n]
__global__ void scan_kernel(const int* __restrict__ cnt, int* __restrict__ rp, int n) {
  __shared__ int sh[1024];
  int base = 0;
  for (int start = 0; start < n; start += 1024) {
    int i = start + (int)threadIdx.x;
    int v = (i < n) ? cnt[i] : 0;
    sh[threadIdx.x] = v;
    __syncthreads();
    for (int off = 1; off < 1024; off <<= 1) {
      int t = (threadIdx.x >= (unsigned)off) ? sh[threadIdx.x - off] : 0;
      __syncthreads();
      sh[threadIdx.x] += t;
      __syncthreads();
    }
    if (i < n) rp[i] = base + sh[threadIdx.x] - v;  // exclusive
    base += sh[1023];
    __syncthreads();
  }
  if (threadIdx.x == 0) rp[n] = base;
}

__global__ void deg_kernel(const int* __restrict__ cnt, float* __restrict__ deg, int n) {
  int i = blockIdx.x * blockDim.x + threadIdx.x;
  if (i < n) deg[i] = fmaxf((float)cnt[i], 1.0f);
}

__global__ void copy_i32_kernel(const int* __restrict__ a, int* __restrict__ b, int n) {
  int i = blockIdx.x * blockDim.x + threadIdx.x;
  if (i < n) b[i] = a[i];
}

__global__ void edge_scatter_kernel(const int* __restrict__ e0, const int* __restrict__ e1,
                                    int* __restrict__ fill, int* __restrict__ col) {
  int i = blockIdx.x * blockDim.x + threadIdx.x;
  if (i < EE) {
    int r = e0[i];
    int p = atomicAdd(&fill[r], 1);
    col[p] = e1[i];
  }
}

// ---------------- embedding gather ----------------
__global__ void emb_gather_kernel(const float* __restrict__ emb, const int* __restrict__ nt,
                                  float* __restrict__ x) {
  int t = blockIdx.x * blockDim.x + threadIdx.x;
  if (t < NN * 64) {
    int i = t >> 6, c = t & 63;
    x[t] = emb[nt[i] * 64 + c];
  }
}

// ---------------- SAGE mean aggregation: agg[r] = (sum_e x[col[e]]) / deg[r] ----------------
__global__ void agg_kernel(const float* __restrict__ xin, int ldx,
                           const int* __restrict__ rp, const int* __restrict__ col,
                           const float* __restrict__ deg, float* __restrict__ agg) {
  int r = blockIdx.x;
  int c = threadIdx.x;         // 64 channels
  int s = rp[r], e = rp[r + 1];
  float acc = 0.f;
  for (int j = s; j < e; ++j) acc += xin[col[j] * ldx + c];
  agg[r * 64 + c] = acc / deg[r];
}

// ---------------- fused linear via fp32 WMMA: O = [relu]( A1@W1^T [+ A2@W2^T] + b ) ----------------
// Weights are [Cout, Cin] row-major, staged zero-padded in LDS (no divergence in the
// WMMA loop: B fragments come from ds_load_b64, A fragments from global_load_b64).
// One wave -> one 16x16 output tile; 8 waves/block share the staged weights.
__global__ void wmma_linear_kernel(const float* __restrict__ A1, int ld1, const float* __restrict__ W1,
                                   const float* __restrict__ A2, int ld2, const float* __restrict__ W2,
                                   const float* __restrict__ bias, float* __restrict__ O, int ldo,
                                   int M, int Cin, int Cout, int do_relu) {
  __shared__ float sw[8192];           // 32 KB: fits padded W1 [+ W2] for all call sites
  int nT = (Cout + 15) >> 4;
  int padR = nT << 4;
  int tot = padR * Cin;

  for (int t = threadIdx.x; t < tot; t += blockDim.x) {
    int r = t / Cin, c = t - r * Cin;
    sw[t] = (r < Cout) ? W1[r * Cin + c] : 0.f;
  }
  if (A2) {
    for (int t = threadIdx.x; t < tot; t += blockDim.x) {
      int r = t / Cin, c = t - r * Cin;
      sw[tot + t] = (r < Cout) ? W2[r * Cin + c] : 0.f;
    }
  }
  __syncthreads();

  int wave = (int)((blockIdx.x * blockDim.x + threadIdx.x) >> 5);
  int lane = threadIdx.x & 31;
  int mT = M >> 4;
  if (wave >= mT * nT) return;         // wave-uniform: EXEC stays all-ones for WMMA
  int m0 = (wave / nT) << 4;
  int n0 = (wave % nT) << 4;
  int lo = lane & 15;
  int hi = lane >> 4;                  // 0 or 1
  int arow = m0 + lo;                  // M = 6000 is an exact multiple of 16
  int koff = 2 * hi;                   // lane-half K offset

  const float* wb1 = &sw[(n0 + lo) * Cin];
  v8f acc = {0.f, 0.f, 0.f, 0.f, 0.f, 0.f, 0.f, 0.f};

  for (int k = 0; k < Cin; k += 4) {
    // A 16x4 fp32 fragment: lanes 0-15 -> K=k..k+1, lanes 16-31 -> K=k+2..k+3
    v2f a = *(const v2f*)&A1[arow * ld1 + k + koff];
    // B 4x16 fragment (B = W^T): N = n0+lo, same K split across lane halves
    v2f b = *(const v2f*)&wb1[k + koff];
    acc = __builtin_amdgcn_wmma_f32_16x16x4_f32(false, a, false, b, (short)0, acc, false, false);
  }
  if (A2) {
    const float* wb2 = &sw[tot + (n0 + lo) * Cin];
    for (int k = 0; k < Cin; k += 4) {
      v2f a = *(const v2f*)&A2[arow * ld2 + k + koff];
      v2f b = *(const v2f*)&wb2[k + koff];
      acc = __builtin_amdgcn_wmma_f32_16x16x4_f32(false, a, false, b, (short)0, acc, false, false);
    }
  }

  int colN = n0 + lo;
  if (colN < Cout) {
    float bv = bias[colN];
#pragma unroll
    for (int v = 0; v < 8; ++v) {
      int row = m0 + v + 8 * hi;       // D layout: VGPR v -> M = v (lo half) / v+8 (hi half)
      float y = acc[v] + bv;
      if (do_relu) y = fmaxf(y, 0.f);
      O[row * ldo + colN] = y;
    }
  }
}

// ---------------- BatchNorm (training mode, biased var), deterministic reductions ----------------
__global__ void bn_stats_kernel(const float* __restrict__ p, int ld, int C, float* __restrict__ accum) {
  __shared__ float ss[256], sq[256];
  int c = blockIdx.x;                  // one block per channel
  int t = threadIdx.x;
  float s = 0.f, q = 0.f;
  for (int r = t; r < NN; r += 256) {
    float v = p[r * ld + c];
    s += v; q += v * v;
  }
  ss[t] = s; sq[t] = q; __syncthreads();
  for (int o = 128; o > 0; o >>= 1) {
    if (t < o) { ss[t] += ss[t + o]; sq[t] += sq[t + o]; }
    __syncthreads();
  }
  if (t == 0) { accum[c] = ss[0]; accum[C + c] = sq[0]; }
}

__global__ void bn_apply_kernel(float* __restrict__ p, int ld, int C, const float* __restrict__ accum) {
  int t = blockIdx.x * blockDim.x + threadIdx.x;
  if (t >= NN * C) return;
  int r = t / C, c = t % C;
  float m = accum[c] * (1.0f / NN);
  float v = accum[C + c] * (1.0f / NN) - m * m;
  p[r * ld + c] = (p[r * ld + c] - m) * rsqrtf(v + 1e-5f);
}

// ---------------- pooled features: sx[k][c] = sum_i softmax(s[i])[k] * xe[i][c] ----------------
__global__ void sx_kernel(const float* __restrict__ s, const float* __restrict__ xe,
                          float* __restrict__ sx) {
  __shared__ float red[256];
  int k = blockIdx.x / 192, c = blockIdx.x % 192;
  int t = threadIdx.x;
  float acc = 0.f;
  for (int i = t; i < NN; i += 256) {
    float s0 = s[i * 4 + 0], s1 = s[i * 4 + 1], s2 = s[i * 4 + 2], s3 = s[i * 4 + 3];
    float m = fmaxf(fmaxf(s0, s1), fmaxf(s2, s3));
    float e0 = expf(s0 - m), e1 = expf(s1 - m), e2 = expf(s2 - m), e3 = expf(s3 - m);
    float inv = 1.f / (e0 + e1 + e2 + e3);
    float w = ((k == 0) ? e0 : (k == 1) ? e1 : (k == 2) ? e2 : e3) * inv;
    acc += w * xe[i * 192 + c];
  }
  red[t] = acc; __syncthreads();
  for (int o = 128; o > 0; o >>= 1) {
    if (t < o) red[t] += red[t + o];
    __syncthreads();
  }
  if (t == 0) sx[k * 192 + c] = red[0];
}

// ---------------- tiny [4,*] linears ----------------
__global__ void lin1_kernel(const float* __restrict__ sx, const float* __restrict__ w,
                            const float* __restrict__ b, float* __restrict__ o) {
  int t = threadIdx.x;                 // 256 = 4*64
  int k = t >> 6, j = t & 63;
  float acc = b[j];
  for (int c = 0; c < 192; ++c) acc += sx[k * 192 + c] * w[j * 192 + c];
  o[k * 64 + j] = fmaxf(acc, 0.f);
}

__global__ void lin2_kernel(const float* __restrict__ sx1, const float* __restrict__ w,
                            const float* __restrict__ b, float* __restrict__ o) {
  int t = threadIdx.x;                 // 64 = 4*16
  int k = t >> 4, j = t & 15;
  float acc = b[j];
  for (int c = 0; c < 64; ++c) acc += sx1[k * 64 + c] * w[j * 64 + c];
  o[k * 16 + j] = acc;
}

__global__ void argmax_kernel(const float* __restrict__ s, int* __restrict__ idx) {
  int i = blockIdx.x * blockDim.x + threadIdx.x;
  if (i < NN) {
    float best = s[i * 4]; int bi = 0;
    for (int k = 1; k < 4; ++k) { float v = s[i * 4 + k]; if (v > best) { best = v; bi = k; } }
    idx[i] = bi;
  }
}

__global__ void gather_res_kernel(const float* __restrict__ sx2, const int* __restrict__ idx,
                                  float* __restrict__ cat) {
  int t = blockIdx.x * blockDim.x + threadIdx.x;
  if (t < NN * 16) {
    int i = t >> 4, j = t & 15;
    cat[i * 48 + 32 + j] = sx2[idx[i] * 16 + j];
  }
}

// ---------------- host-side orchestration ----------------
extern "C" void kernel_launch(void* const* d_in, const int* in_sizes, int n_in,
                              void* d_out, int out_size, void* d_ws, size_t ws_size,
                              hipStream_t stream) {
  const float* emb_w  = (const float*)d_in[0];
  const float* pw1r   = (const float*)d_in[1];
  const float* pw1l   = (const float*)d_in[2];
  const float* pb1    = (const float*)d_in[3];
  const float* pw2r   = (const float*)d_in[4];
  const float* pw2l   = (const float*)d_in[5];
  const float* pb2    = (const float*)d_in[6];
  const float* pw3r   = (const float*)d_in[7];
  const float* pw3l   = (const float*)d_in[8];
  const float* pb3    = (const float*)d_in[9];
  const float* plin_w = (const float*)d_in[10];
  const float* plin_b = (const float*)d_in[11];
  const float* ew1r   = (const float*)d_in[12];
  const float* ew1l   = (const float*)d_in[13];
  const float* eb1    = (const float*)d_in[14];
  const float* ew2r   = (const float*)d_in[15];
  const float* ew2l   = (const float*)d_in[16];
  const float* eb2    = (const float*)d_in[17];
  const float* ew3r   = (const float*)d_in[18];
  const float* ew3l   = (const float*)d_in[19];
  const float* eb3    = (const float*)d_in[20];
  const float* lin1_w = (const float*)d_in[21];
  const float* lin1_b = (const float*)d_in[22];
  const float* lin2_w = (const float*)d_in[23];
  const float* lin2_b = (const float*)d_in[24];
  const float* embl_w = (const float*)d_in[25];
  const float* embl_b = (const float*)d_in[26];
  const float* lin3_w = (const float*)d_in[27];
  const float* lin3_b = (const float*)d_in[28];
  const int*   e0     = (const int*)d_in[29];        // edge_index row 0
  const int*   e1     = ((const int*)d_in[29]) + EE; // edge_index row 1
  const int*   node_t = (const int*)d_in[30];

  // workspace layout (~13 MB)
  float* ws   = (float*)d_ws;
  float* x    = ws;                 // [6000,64]
  float* xcp  = x    + NN * 64;     // [6000,132] pool concat
  float* xce  = xcp  + NN * 132;    // [6000,192] embed concat
  float* agg  = xce  + NN * 192;    // [6000,64]
  float* deg  = agg  + NN * 64;     // [6000]
  float* sbuf = deg  + NN;          // [6000,4]
  float* cat  = sbuf + NN * 4;      // [6000,48] = x_embed(32) ++ res(16)
  float* sx   = cat  + NN * 48;     // [4,192]
  float* sx1  = sx   + 768;         // [4,64]
  float* sx2  = sx1  + 256;         // [4,16]
  float* bnacc = sx2 + 64;          // [2*64]
  int* row_count = (int*)(bnacc + 128); // [6000]
  int* row_ptr   = row_count + NN;      // [6001]
  int* row_fill  = row_ptr + NN + 4;    // [6000]
  int* csr_col   = row_fill + NN;       // [192000]
  int* idxb      = csr_col + EE;        // [6000]

  // --- sparse adjacency -> CSR (counting sort) ---
  zero_i32_kernel<<<(NN + 255) / 256, 256, 0, stream>>>(row_count, NN);
  edge_count_kernel<<<(EE + 255) / 256, 256, 0, stream>>>(e0, row_count);
  scan_kernel<<<1, 1024, 0, stream>>>(row_count, row_ptr, NN);
  deg_kernel<<<(NN + 255) / 256, 256, 0, stream>>>(row_count, deg, NN);
  copy_i32_kernel<<<(NN + 255) / 256, 256, 0, stream>>>(row_ptr, row_fill, NN);
  edge_scatter_kernel<<<(EE + 255) / 256, 256, 0, stream>>>(e0, e1, row_fill, csr_col);

  emb_gather_kernel<<<(NN * 64 + 255) / 256, 256, 0, stream>>>(emb_w, node_t, x);

  auto launch_lin = [&](const float* A1, int ld1, const float* W1,
                        const float* A2, int ld2, const float* W2,
                        const float* bias, float* O, int ldo, int Cin, int Cout, int relu) {
    int nT = (Cout + 15) / 16;
    int waves = (NN / 16) * nT;
    int blocks = (waves + 7) / 8;   // 256 threads = 8 waves per block
    wmma_linear_kernel<<<blocks, 256, 0, stream>>>(A1, ld1, W1, A2, ld2, W2,
                                                   bias, O, ldo, NN, Cin, Cout, relu);
  };

  auto sage = [&](const float* xin, int ldin, const float* wr, const float* wl,
                  const float* bias, float* outp, int ldo, int Cout) {
    agg_kernel<<<NN, 64, 0, stream>>>(xin, ldin, row_ptr, csr_col, deg, agg);
    launch_lin(agg, 64, wr, xin, ldin, wl, bias, outp, ldo, 64, Cout, 1); // relu(sage)
    bn_stats_kernel<<<Cout, 256, 0, stream>>>(outp, ldo, Cout, bnacc);
    bn_apply_kernel<<<(NN * Cout + 255) / 256, 256, 0, stream>>>(outp, ldo, Cout, bnacc);
  };

  // pool GNN -> s
  sage(x,        64,  pw1r, pw1l, pb1, xcp + 0,   132, 64);
  sage(xcp + 0,  132, pw2r, pw2l, pb2, xcp + 64,  132, 64);
  sage(xcp + 64, 132, pw3r, pw3l, pb3, xcp + 128, 132, 4);
  launch_lin(xcp, 132, plin_w, nullptr, 0, nullptr, plin_b, sbuf, 4, 132, 4, 1);

  // embed GNN -> xe
  sage(x,        64,  ew1r, ew1l, eb1, xce + 0,   192, 64);
  sage(xce + 0,  192, ew2r, ew2l, eb2, xce + 64,  192, 64);
  sage(xce + 64, 192, ew3r, ew3l, eb3, xce + 128, 192, 64);

  // x_embed -> cat[:, 0:32]
  launch_lin(xce, 192, embl_w, nullptr, 0, nullptr, embl_b, cat, 48, 192, 32, 0);

  // pooled features + two tiny linears
  sx_kernel<<<4 * 192, 256, 0, stream>>>(sbuf, xce, sx);
  lin1_kernel<<<1, 256, 0, stream>>>(sx, lin1_w, lin1_b, sx1);
  lin2_kernel<<<1, 64, 0, stream>>>(sx1, lin2_w, lin2_b, sx2);

  // per-node cluster gather -> cat[:, 32:48]
  argmax_kernel<<<(NN + 255) / 256, 256, 0, stream>>>(sbuf, idxb);
  gather_res_kernel<<<(NN * 16 + 255) / 256, 256, 0, stream>>>(sx2, idxb, cat);

  // final linear -> d_out [6000,16]
  launch_lin(cat, 48, lin3_w, nullptr, 0, nullptr, lin3_b, (float*)d_out, 16, 48, 16, 0);
}